// FraudGNN_85143431676315
// MI455X (gfx1250) — compile-verified
//
#include <hip/hip_runtime.h>
#include <hip/hip_bf16.h>

typedef __attribute__((ext_vector_type(2))) float v2f;
typedef __attribute__((ext_vector_type(8))) float v8f;

// ---------------- degree / normalization ----------------
__global__ void deg_init(float* __restrict__ deg, int N) {
    int i = blockIdx.x * blockDim.x + threadIdx.x;
    if (i < N) deg[i] = 1.0f;  // self-loop contributes 1
}

__global__ void deg_count(const int* __restrict__ dst, float* __restrict__ deg, int E) {
    int e = blockIdx.x * blockDim.x + threadIdx.x;
    if (e < E) atomicAdd(&deg[dst[e]], 1.0f);
}

__global__ void deg_rsqrt(float* __restrict__ deg, int N) {
    int i = blockIdx.x * blockDim.x + threadIdx.x;
    if (i < N) deg[i] = rsqrtf(deg[i]);  // deg >= 1 always
}

// ---------------- GEMM1: h1 = x @ W1  (Nx8 @ 8x32), fp32 WMMA 16x16x4 ----------------
// One wave per 16-row tile. Also initializes agg1 with the self-loop message
// h1[i] * dinv[i]^2 so the edge-scatter kernel only adds edge contributions.
__global__ __launch_bounds__(256) void gemm1_wmma(
    const float* __restrict__ x, const float* __restrict__ W1,
    const float* __restrict__ dinv,
    float* __restrict__ h1, float* __restrict__ agg1, int N)
{
    int wave = (blockIdx.x * blockDim.x + threadIdx.x) >> 5;
    int lane = threadIdx.x & 31;
    int row0 = wave * 16;
    if (row0 >= N) return;                 // uniform per wave: EXEC stays all-1s
    int half = lane >> 4;                  // 0: K0/K1 & D rows 0-7 ; 1: K2/K3 & D rows 8-15
    int n    = lane & 15;
    int rowA = row0 + (lane & 15);
    int rA   = rowA < N ? rowA : N - 1;    // clamp loads; stores handled below

    v8f c0 = {};                           // N-tile 0 (cols 0..15)
    v8f c1 = {};                           // N-tile 1 (cols 16..31)
#pragma unroll
    for (int kk = 0; kk < 2; ++kk) {       // K = 8 -> two 16x16x4 steps
        int kb = kk * 4 + half * 2;
        v2f a;
        a.x = x[(size_t)rA * 8 + kb];
        a.y = x[(size_t)rA * 8 + kb + 1];
        v2f b0, b1v;
        b0.x  = W1[(size_t)kb * 32 + n];
        b0.y  = W1[(size_t)(kb + 1) * 32 + n];
        b1v.x = W1[(size_t)kb * 32 + 16 + n];
        b1v.y = W1[(size_t)(kb + 1) * 32 + 16 + n];
        c0 = __builtin_amdgcn_wmma_f32_16x16x4_f32(false, a, false, b0,  (short)0, c0, false, false);
        c1 = __builtin_amdgcn_wmma_f32_16x16x4_f32(false, a, false, b1v, (short)0, c1, false, false);
    }
    if (row0 + 16 <= N) {                  // fast path: full tile, branch-free stores
#pragma unroll
        for (int r = 0; r < 8; ++r) {
            int row   = row0 + r + half * 8;
            float dv  = dinv[row];
            float dv2 = dv * dv;
            size_t o  = (size_t)row * 32 + n;
            h1[o]        = c0[r];
            agg1[o]      = c0[r] * dv2;
            h1[o + 16]   = c1[r];
            agg1[o + 16] = c1[r] * dv2;
        }
    } else {                               // tail tile: guarded
#pragma unroll
        for (int r = 0; r < 8; ++r) {
            int row = row0 + r + half * 8;
            if (row < N) {
                float dv  = dinv[row];
                float dv2 = dv * dv;
                size_t o  = (size_t)row * 32 + n;
                h1[o]        = c0[r];
                agg1[o]      = c0[r] * dv2;
                h1[o + 16]   = c1[r];
                agg1[o + 16] = c1[r] * dv2;
            }
        }
    }
}

// ---------------- GEMM2: h2 = relu(agg1 + b1) @ W2  (Nx32 @ 32x16) ----------------
__global__ __launch_bounds__(256) void gemm2_wmma(
    const float* __restrict__ agg1, const float* __restrict__ b1,
    const float* __restrict__ W2, const float* __restrict__ dinv,
    float* __restrict__ h2, float* __restrict__ agg2, int N)
{
    int wave = (blockIdx.x * blockDim.x + threadIdx.x) >> 5;
    int lane = threadIdx.x & 31;
    int row0 = wave * 16;
    if (row0 >= N) return;
    int half = lane >> 4;
    int n    = lane & 15;
    int rowA = row0 + (lane & 15);
    int rA   = rowA < N ? rowA : N - 1;

    v8f c = {};
#pragma unroll
    for (int kk = 0; kk < 8; ++kk) {       // K = 32 -> eight 16x16x4 steps
        int kb = kk * 4 + half * 2;
        float ax = agg1[(size_t)rA * 32 + kb]     + b1[kb];
        float ay = agg1[(size_t)rA * 32 + kb + 1] + b1[kb + 1];
        v2f a;
        a.x = ax > 0.0f ? ax : 0.0f;       // fused ReLU on the A-load path
        a.y = ay > 0.0f ? ay : 0.0f;
        v2f b;
        b.x = W2[(size_t)kb * 16 + n];
        b.y = W2[(size_t)(kb + 1) * 16 + n];
        c = __builtin_amdgcn_wmma_f32_16x16x4_f32(false, a, false, b, (short)0, c, false, false);
    }
    if (row0 + 16 <= N) {                  // fast path
#pragma unroll
        for (int r = 0; r < 8; ++r) {
            int row   = row0 + r + half * 8;
            float dv  = dinv[row];
            size_t o  = (size_t)row * 16 + n;
            h2[o]   = c[r];
            agg2[o] = c[r] * dv * dv;      // self-loop init
        }
    } else {
#pragma unroll
        for (int r = 0; r < 8; ++r) {
            int row = row0 + r + half * 8;
            if (row < N) {
                float dv  = dinv[row];
                size_t o  = (size_t)row * 16 + n;
                h2[o]   = c[r];
                agg2[o] = c[r] * dv * dv;
            }
        }
    }
}

// ---------------- edge scatter, layer 1: 32 features = 32 lanes, 1 edge / wave ----------------
__global__ __launch_bounds__(256) void scatter32(
    const int* __restrict__ src, const int* __restrict__ dst,
    const float* __restrict__ dinv, const float* __restrict__ h1,
    float* __restrict__ agg1, int E)
{
    int e    = (blockIdx.x * blockDim.x + threadIdx.x) >> 5;
    int lane = threadIdx.x & 31;
    if (e >= E) return;
    int s = src[e];
    int d = dst[e];
    float norm = dinv[s] * dinv[d];
    float v = h1[(size_t)s * 32 + lane] * norm;   // coalesced 128B gather
    atomicAdd(&agg1[(size_t)d * 32 + lane], v);   // global_atomic_add_f32, no return
}

// ---------------- edge scatter, layer 2: 16 features, 2 edges / wave ----------------
__global__ __launch_bounds__(256) void scatter16(
    const int* __restrict__ src, const int* __restrict__ dst,
    const float* __restrict__ dinv, const float* __restrict__ h2,
    float* __restrict__ agg2, int E)
{
    int wave = (blockIdx.x * blockDim.x + threadIdx.x) >> 5;
    int lane = threadIdx.x & 31;
    int half = lane >> 4;
    int j    = lane & 15;
    long e   = (long)wave * 2 + half;
    if (e >= E) return;
    int s = src[e];
    int d = dst[e];
    float norm = dinv[s] * dinv[d];
    float v = h2[(size_t)s * 16 + j] * norm;
    atomicAdd(&agg2[(size_t)d * 16 + j], v);
}

// ---------------- head: sigmoid(relu(agg2 + b2) @ Wfc + bfc) ----------------
__global__ __launch_bounds__(256) void head_k(
    const float* __restrict__ agg2, const float* __restrict__ b2,
    const float* __restrict__ Wfc, const float* __restrict__ bfc,
    float* __restrict__ out, int N)
{
    int i = blockIdx.x * blockDim.x + threadIdx.x;
    if (i >= N) return;
    const float4* row = (const float4*)(agg2 + (size_t)i * 16);  // 16B-aligned by layout
    float acc = bfc[0];
#pragma unroll
    for (int q = 0; q < 4; ++q) {
        float4 v = row[q];                 // global_load_b128
        float z;
        z = v.x + b2[q * 4 + 0]; acc += (z > 0.0f ? z : 0.0f) * Wfc[q * 4 + 0];
        z = v.y + b2[q * 4 + 1]; acc += (z > 0.0f ? z : 0.0f) * Wfc[q * 4 + 1];
        z = v.z + b2[q * 4 + 2]; acc += (z > 0.0f ? z : 0.0f) * Wfc[q * 4 + 2];
        z = v.w + b2[q * 4 + 3]; acc += (z > 0.0f ? z : 0.0f) * Wfc[q * 4 + 3];
    }
    out[i] = 1.0f / (1.0f + __expf(-acc));
}

extern "C" void kernel_launch(void* const* d_in, const int* in_sizes, int n_in,
                              void* d_out, int out_size, void* d_ws, size_t ws_size,
                              hipStream_t stream)
{
    const float* x   = (const float*)d_in[0];
    const int*   ei  = (const int*)d_in[1];
    const float* W1  = (const float*)d_in[2];
    const float* b1  = (const float*)d_in[3];
    const float* W2  = (const float*)d_in[4];
    const float* b2  = (const float*)d_in[5];
    const float* Wfc = (const float*)d_in[6];
    const float* bfc = (const float*)d_in[7];

    int N = in_sizes[0] / 8;
    int E = in_sizes[1] / 2;
    const int* src = ei;
    const int* dst = ei + E;

    // workspace: deg/dinv [N] | h1 [32N] | agg1 [32N] | h2 [16N] | agg2 [16N]
    // (each region is 16B-aligned when N is a multiple of 4)
    float* deg  = (float*)d_ws;
    float* h1   = deg  + N;
    float* agg1 = h1   + (size_t)N * 32;
    float* h2   = agg1 + (size_t)N * 32;
    float* agg2 = h2   + (size_t)N * 16;

    const int T = 256;
    deg_init <<<(N + T - 1) / T, T, 0, stream>>>(deg, N);
    deg_count<<<(E + T - 1) / T, T, 0, stream>>>(dst, deg, E);
    deg_rsqrt<<<(N + T - 1) / T, T, 0, stream>>>(deg, N);

    int tiles   = (N + 15) / 16;       // one wave per 16-row tile
    int gblocks = (tiles + 7) / 8;     // 8 waves per 256-thread block
    gemm1_wmma<<<gblocks, T, 0, stream>>>(x, W1, deg, h1, agg1, N);

    long waves1 = E;                   // one wave per edge
    scatter32<<<(int)((waves1 + 7) / 8), T, 0, stream>>>(src, dst, deg, h1, agg1, E);

    gemm2_wmma<<<gblocks, T, 0, stream>>>(agg1, b1, W2, deg, h2, agg2, N);

    long waves2 = ((long)E + 1) / 2;   // two edges per wave
    scatter16<<<(int)((waves2 + 7) / 8), T, 0, stream>>>(src, dst, deg, h2, agg2, E);

    head_k<<<(N + T - 1) / T, T, 0, stream>>>(agg2, b2, Wfc, bfc, (float*)d_out, N);
}